// FREEDOMModel_26465588478613
// MI455X (gfx1250) — compile-verified
//
#include <hip/hip_runtime.h>
#include <hip/hip_bf16.h>
#include <stdint.h>

#define UCNT   100000
#define ICNT   50000
#define KDIM   128
#define NNZ_MM 500000
#define E_UI   1000000
#define NTOT   (UCNT + ICNT)
#define EPSF   1e-7f

// ---------------- helpers ----------------

__device__ __forceinline__ void atom_add_f32(float* p, float v) {
    // relaxed, device scope -> global_atomic_add_f32 (no return)
    __hip_atomic_fetch_add(p, v, __ATOMIC_RELAXED, __HIP_MEMORY_SCOPE_AGENT);
}

// CDNA5 async global->LDS copy, 16B per lane. ldsoff is the wave-relative LDS
// byte address (low 32 bits of the generic pointer == LDS offset per ISA 10.2).
__device__ __forceinline__ void async_ld16(const float* g, unsigned ldsoff) {
    asm volatile("global_load_async_to_lds_b128 %0, %1, off"
                 :: "v"(ldsoff), "v"(g)
                 : "memory");
}

// ---------------- elementwise kernels ----------------

__global__ void k_zero(float* __restrict__ p, int n) {
    int i = blockIdx.x * blockDim.x + threadIdx.x;
    if (i < n) p[i] = 0.0f;
}

__global__ void k_zero4(float4* __restrict__ p, int n4) {
    int i = blockIdx.x * blockDim.x + threadIdx.x;
    if (i < n4) p[i] = make_float4(0.f, 0.f, 0.f, 0.f);
}

// deg <- rsqrt(eps + deg), in place (applied to all three degree arrays at once)
__global__ void k_rsqrt(float* __restrict__ p, int n) {
    int i = blockIdx.x * blockDim.x + threadIdx.x;
    if (i < n) p[i] = rsqrtf(EPSF + p[i]);
}

// ego0 = concat(Gu, Gi)
__global__ void k_init(const float4* __restrict__ Gu, const float4* __restrict__ Gi,
                       float4* __restrict__ ego) {
    int i = blockIdx.x * blockDim.x + threadIdx.x;  // over NTOT*K/4
    const int n4 = NTOT * KDIM / 4;
    const int u4 = UCNT * KDIM / 4;
    if (i < n4) {
        ego[i] = (i < u4) ? Gu[i] : Gi[i - u4];
    }
}

// out = (concat(Gu,Gi) + ego1 + ego2) / 3   (single fused sweep)
__global__ void k_final(const float4* __restrict__ Gu, const float4* __restrict__ Gi,
                        const float4* __restrict__ ego1, const float4* __restrict__ ego2,
                        float4* __restrict__ out) {
    int i = blockIdx.x * blockDim.x + threadIdx.x;
    const int n4 = NTOT * KDIM / 4;
    const int u4 = UCNT * KDIM / 4;
    if (i < n4) {
        float4 v = (i < u4) ? Gu[i] : Gi[i - u4];
        float4 a = ego1[i];
        float4 b = ego2[i];
        const float s = 1.0f / 3.0f;
        float4 r;
        r.x = (v.x + a.x + b.x) * s;
        r.y = (v.y + a.y + b.y) * s;
        r.z = (v.z + a.z + b.z) * s;
        r.w = (v.w + a.w + b.w) * s;
        out[i] = r;
    }
}

// ---------------- degree counting ----------------

__global__ void k_deg_mm(const int* __restrict__ rows, float* __restrict__ deg) {
    int e = blockIdx.x * blockDim.x + threadIdx.x;
    if (e < NNZ_MM) atom_add_f32(&deg[rows[e]], 1.0f);
}

__global__ void k_deg_ui(const int* __restrict__ rows, const int* __restrict__ cols,
                         float* __restrict__ udeg, float* __restrict__ ideg) {
    int e = blockIdx.x * blockDim.x + threadIdx.x;
    if (e < E_UI) {
        atom_add_f32(&udeg[rows[e]], 1.0f);
        atom_add_f32(&ideg[cols[e]], 1.0f);
    }
}

// ---------------- UI bipartite SpMM (symmetrized), async-pipelined ----------------
// One wave per edge (grid-stride). Edge (u,i), val = rsq_u[u]*rsq_i[i] (precomputed):
//   next[u]   += val * cur[U+i]
//   next[U+i] += val * cur[u]
// The two 512B source rows of the NEXT edge are staged into LDS via the CDNA5
// async copy engine while the current edge's atomics are issued (double-buffered,
// ASYNCcnt-tracked; async loads complete in order so waiting <=2 releases the
// current slot only).

#define SP_WAVES 4
#define SP_TPB   (SP_WAVES * 32)

__global__ __launch_bounds__(SP_TPB)
void k_spmm_ui(const int* __restrict__ rows, const int* __restrict__ cols,
               const float* __restrict__ ursq, const float* __restrict__ irsq,
               const float* __restrict__ cur, float* __restrict__ next) {
    __shared__ __align__(16) float stage[SP_WAVES * 2 /*slots*/ * 2 /*rows*/ * KDIM];

    const int  lane = threadIdx.x & 31;
    const int  wv   = threadIdx.x >> 5;
    const long wid  = (long)blockIdx.x * SP_WAVES + wv;
    const long nw   = (long)gridDim.x * SP_WAVES;

    float*   base     = &stage[wv * (2 * 2 * KDIM)];
    unsigned base_off = (unsigned)(uintptr_t)base;   // LDS byte offset of this wave's staging

    // prologue: stage edge `wid` into slot 0
    long e = wid;
    if (e < E_UI) {
        int u = rows[e], it = cols[e];
        async_ld16(&cur[(size_t)u * KDIM + lane * 4],
                   base_off + (unsigned)((0 * KDIM + lane * 4) * 4));
        async_ld16(&cur[(size_t)(UCNT + it) * KDIM + lane * 4],
                   base_off + (unsigned)((1 * KDIM + lane * 4) * 4));
    }

    int slot = 0;
    for (; e < E_UI; e += nw, slot ^= 1) {
        long en = e + nw;
        if (en < E_UI) {
            // stage next edge into the other slot, then wait for current slot only
            int u2 = rows[en], i2 = cols[en];
            unsigned so = base_off + (unsigned)(((slot ^ 1) * 2 * KDIM) * 4);
            async_ld16(&cur[(size_t)u2 * KDIM + lane * 4],
                       so + (unsigned)((0 * KDIM + lane * 4) * 4));
            async_ld16(&cur[(size_t)(UCNT + i2) * KDIM + lane * 4],
                       so + (unsigned)((1 * KDIM + lane * 4) * 4));
            asm volatile("s_wait_asynccnt 2" ::: "memory");
        } else {
            asm volatile("s_wait_asynccnt 0" ::: "memory");
        }

        // process edge `e` from the current slot
        int u = rows[e], it = cols[e];
        float val = ursq[u] * irsq[it];

        float4 xu = *(const float4*)&base[(slot * 2 + 0) * KDIM + lane * 4]; // cur[u]
        float4 xi = *(const float4*)&base[(slot * 2 + 1) * KDIM + lane * 4]; // cur[U+i]

        float* ou = &next[(size_t)u * KDIM + lane * 4];           // += val*cur[U+i]
        float* oi = &next[(size_t)(UCNT + it) * KDIM + lane * 4]; // += val*cur[u]
        atom_add_f32(ou + 0, val * xi.x);
        atom_add_f32(ou + 1, val * xi.y);
        atom_add_f32(ou + 2, val * xi.z);
        atom_add_f32(ou + 3, val * xi.w);
        atom_add_f32(oi + 0, val * xu.x);
        atom_add_f32(oi + 1, val * xu.y);
        atom_add_f32(oi + 2, val * xu.z);
        atom_add_f32(oi + 3, val * xu.w);
    }
}

// ---------------- item-item SpMM, h added directly into out[U+..] ----------------

__global__ __launch_bounds__(256)
void k_spmm_mm(const int* __restrict__ rows, const int* __restrict__ cols,
               const float* __restrict__ rsq, const float* __restrict__ Gi,
               float* __restrict__ out) {
    const int  lane = threadIdx.x & 31;
    const int  wv   = threadIdx.x >> 5;
    const long wid  = (long)blockIdx.x * (blockDim.x / 32) + wv;
    const long nw   = (long)gridDim.x * (blockDim.x / 32);

    for (long e = wid; e < NNZ_MM; e += nw) {
        long en = e + nw;
        if (en < NNZ_MM) {   // prefetch next source row (global_prefetch_b8)
            __builtin_prefetch(&Gi[(size_t)cols[en] * KDIM + lane * 4], 0, 0);
        }
        int r = rows[e], c = cols[e];
        float val = rsq[r] * rsq[c];
        float4 g = *(const float4*)&Gi[(size_t)c * KDIM + lane * 4];
        float* o = &out[(size_t)(UCNT + r) * KDIM + lane * 4];
        atom_add_f32(o + 0, val * g.x);
        atom_add_f32(o + 1, val * g.y);
        atom_add_f32(o + 2, val * g.z);
        atom_add_f32(o + 3, val * g.w);
    }
}

// ---------------- driver ----------------

extern "C" void kernel_launch(void* const* d_in, const int* in_sizes, int n_in,
                              void* d_out, int out_size, void* d_ws, size_t ws_size,
                              hipStream_t stream) {
    const float* Gu   = (const float*)d_in[0];
    const float* Gi   = (const float*)d_in[1];
    const int*   mm   = (const int*)d_in[2];      // [2, NNZ_MM] -> rows then cols
    const int*   ui_r = (const int*)d_in[3];
    const int*   ui_c = (const int*)d_in[4];
    float*       out  = (float*)d_out;

    float* ws     = (float*)d_ws;
    float* mm_deg = ws;                              // I   (becomes rsqrt values)
    float* u_deg  = mm_deg + ICNT;                   // U   (becomes rsqrt values)
    float* i_deg  = u_deg + UCNT;                    // I   (becomes rsqrt values)
    float* ego_a  = i_deg + ICNT;                    // N*K (ego0, later ego2)
    float* ego_b  = ego_a + (size_t)NTOT * KDIM;     // N*K (ego1)

    const int T  = 256;
    const int n4 = NTOT * KDIM / 4;
    const int ndeg = ICNT + UCNT + ICNT;

    // degrees -> rsqrt(eps + deg)
    k_zero  <<<(ndeg + T - 1) / T, T, 0, stream>>>(mm_deg, ndeg);
    k_deg_mm<<<(NNZ_MM + T - 1) / T, T, 0, stream>>>(mm, mm_deg);
    k_deg_ui<<<(E_UI + T - 1) / T, T, 0, stream>>>(ui_r, ui_c, u_deg, i_deg);
    k_rsqrt <<<(ndeg + T - 1) / T, T, 0, stream>>>(mm_deg, ndeg);

    // ego0 = concat(Gu, Gi)
    k_init<<<(n4 + T - 1) / T, T, 0, stream>>>((const float4*)Gu, (const float4*)Gi,
                                               (float4*)ego_a);

    // layer 1: ego1 (ego_b) = A @ ego0 (ego_a)
    k_zero4  <<<(n4 + T - 1) / T, T, 0, stream>>>((float4*)ego_b, n4);
    k_spmm_ui<<<3072, SP_TPB, 0, stream>>>(ui_r, ui_c, u_deg, i_deg, ego_a, ego_b);

    // layer 2: ego2 (ego_a, ego0 no longer needed - it equals concat(Gu,Gi)) = A @ ego1
    k_zero4  <<<(n4 + T - 1) / T, T, 0, stream>>>((float4*)ego_a, n4);
    k_spmm_ui<<<3072, SP_TPB, 0, stream>>>(ui_r, ui_c, u_deg, i_deg, ego_b, ego_a);

    // out = (ego0 + ego1 + ego2) / 3, reading ego0 straight from the inputs
    k_final<<<(n4 + T - 1) / T, T, 0, stream>>>((const float4*)Gu, (const float4*)Gi,
                                                (const float4*)ego_b, (const float4*)ego_a,
                                                (float4*)out);

    // i_g += h = mm_norm @ Gi  (added straight into the item half of out)
    k_spmm_mm<<<2048, 256, 0, stream>>>(mm, mm + NNZ_MM, mm_deg, Gi, out);
}